// FastPhysicsEvaluator_42305427865966
// MI455X (gfx1250) — compile-verified
//
#include <hip/hip_runtime.h>
#include <cstdint>
#include <cstddef>

// Problem geometry (fixed by the reference): B=16, C=1, H=W=1024, fp32.
constexpr int   Bn     = 16;
constexpr int   Hh     = 1024;
constexpr int   Ww     = 1024;
constexpr int   ROWS   = 4;                   // rows of output per block
constexpr int   STRIPS = Hh / ROWS;           // 256 strips per image
constexpr int   NBLK   = Bn * STRIPS;         // 4096 blocks
constexpr float NUc    = 0.001f;
constexpr float INV_N  = 1.0f / (16.0f * 1024.0f * 1024.0f);

// Types matching the gfx1250 async-to-LDS builtin signature:
//   void (int4 AS1* gsrc, int4 AS3* ldst, imm offset, imm cpol)
typedef int v4i __attribute__((vector_size(16)));
typedef __attribute__((address_space(1))) v4i GV4;   // global int4
typedef __attribute__((address_space(3))) v4i LV4;   // LDS int4

// ---- CDNA5 async copy: global memory -> LDS, 128 bits per lane ------------
__device__ __forceinline__ void async_load_b128(const float* g, float* l) {
#if __has_builtin(__builtin_amdgcn_global_load_async_to_lds_b128)
  // Route casts through integers to avoid address-space cast diagnostics.
  GV4* gp = (GV4*)(uintptr_t)g;                       // 64-bit global address
  LV4* lp = (LV4*)(uint32_t)(uintptr_t)l;             // low 32 bits = LDS offset
  __builtin_amdgcn_global_load_async_to_lds_b128(gp, lp, /*offset=*/0, /*cpol=*/0);
#else
  uint32_t lds_off = (uint32_t)(uintptr_t)l;
  uint64_t gaddr   = (uint64_t)(uintptr_t)g;
  asm volatile("global_load_async_to_lds_b128 %0, %1, off"
               :: "v"(lds_off), "v"(gaddr)
               : "memory");
#endif
}

__device__ __forceinline__ void async_wait_zero() {
#if __has_builtin(__builtin_amdgcn_s_wait_asynccnt)
  __builtin_amdgcn_s_wait_asynccnt(0);
#else
  asm volatile("s_wait_asynccnt 0x0" ::: "memory");
#endif
}

// ---------------------------------------------------------------------------
// Pass 1: stencil residual + per-block partial sum of (res_x^2 + res_y^2).
// Block = (batch image, 4-row strip). 256 threads, 4 columns per thread.
// ---------------------------------------------------------------------------
__global__ __launch_bounds__(256) void ns_residual_kernel(
    const float* __restrict__ u, const float* __restrict__ v,
    float* __restrict__ partial) {
  __shared__ float su[(ROWS + 2) * Ww];   // 24 KB
  __shared__ float sv[(ROWS + 2) * Ww];   // 24 KB
  __shared__ float wsum[8];

  const int tid   = threadIdx.x;
  const int strip = blockIdx.x % STRIPS;
  const int b     = blockIdx.x / STRIPS;
  const int h0    = strip * ROWS;
  const int c0    = tid << 2;                                // 4 cols/thread
  const size_t base = (size_t)b * (size_t)(Hh * Ww);

  // ---- Stage ROWS+2 rows of u and v into LDS via async-tensor path ----
  #pragma unroll
  for (int r = 0; r < ROWS + 2; ++r) {
    int gr = h0 - 1 + r;                                     // edge replicate
    gr = gr < 0 ? 0 : gr;
    gr = gr > (Hh - 1) ? (Hh - 1) : gr;
    const size_t off = base + (size_t)gr * Ww + (size_t)c0;
    async_load_b128(u + off, &su[r * Ww + c0]);
    async_load_b128(v + off, &sv[r * Ww + c0]);
  }
  async_wait_zero();     // my wave's async copies landed in LDS
  __syncthreads();       // everyone else's too

  // ---- Compute: rolling 3-row register window ----
  const int cl = (c0 == 0) ? 0 : c0 - 1;                     // left halo col
  const int cr = (c0 + 4 > Ww - 1) ? (Ww - 1) : c0 + 4;      // right halo col

  float4 um4 = *(const float4*)&su[0 * Ww + c0];
  float4 uc4 = *(const float4*)&su[1 * Ww + c0];
  float4 vm4 = *(const float4*)&sv[0 * Ww + c0];
  float4 vc4 = *(const float4*)&sv[1 * Ww + c0];

  float acc = 0.0f;
  #pragma unroll
  for (int r = 1; r <= ROWS; ++r) {
    float4 up4 = *(const float4*)&su[(r + 1) * Ww + c0];
    float4 vp4 = *(const float4*)&sv[(r + 1) * Ww + c0];
    const float ulf = su[r * Ww + cl];
    const float urf = su[r * Ww + cr];
    const float vlf = sv[r * Ww + cl];
    const float vrf = sv[r * Ww + cr];

    const float uc[4] = {uc4.x, uc4.y, uc4.z, uc4.w};
    const float um[4] = {um4.x, um4.y, um4.z, um4.w};
    const float up[4] = {up4.x, up4.y, up4.z, up4.w};
    const float vcv[4] = {vc4.x, vc4.y, vc4.z, vc4.w};
    const float vmv[4] = {vm4.x, vm4.y, vm4.z, vm4.w};
    const float vpv[4] = {vp4.x, vp4.y, vp4.z, vp4.w};

    #pragma unroll
    for (int j = 0; j < 4; ++j) {
      const float uL = (j == 0) ? ulf : uc[j - 1];
      const float uR = (j == 3) ? urf : uc[j + 1];
      const float vL = (j == 0) ? vlf : vcv[j - 1];
      const float vR = (j == 3) ? vrf : vcv[j + 1];

      const float ux = (uR - uL) * 0.5f;
      const float uy = (up[j] - um[j]) * 0.5f;
      const float vx = (vR - vL) * 0.5f;
      const float vy = (vpv[j] - vmv[j]) * 0.5f;
      const float lapu = uR + uL + up[j] + um[j] - 4.0f * uc[j];
      const float lapv = vR + vL + vpv[j] + vmv[j] - 4.0f * vcv[j];

      // res_x = u*u_x + v*u_y - nu*lap(u);  res_y = u*v_x + v*v_y - nu*lap(v)
      const float rx = fmaf(uc[j], ux, fmaf(vcv[j], uy, -NUc * lapu));
      const float ry = fmaf(uc[j], vx, fmaf(vcv[j], vy, -NUc * lapv));
      acc = fmaf(rx, rx, acc);
      acc = fmaf(ry, ry, acc);
    }
    um4 = uc4; uc4 = up4;
    vm4 = vc4; vc4 = vp4;
  }

  // ---- Deterministic block reduction (wave32 shuffle, then LDS) ----
  #pragma unroll
  for (int off = 16; off > 0; off >>= 1)
    acc += __shfl_down(acc, off, 32);
  if ((tid & 31) == 0) wsum[tid >> 5] = acc;
  __syncthreads();
  if (tid == 0) {
    float s = 0.0f;
    #pragma unroll
    for (int w = 0; w < 8; ++w) s += wsum[w];
    partial[blockIdx.x] = s;
  }
}

// ---------------------------------------------------------------------------
// Pass 2: sum the 4096 per-block partials, scale by 1/N, write scalar.
// ---------------------------------------------------------------------------
__global__ __launch_bounds__(256) void final_reduce_kernel(
    const float* __restrict__ partial, float* __restrict__ out, int n) {
  __shared__ float wsum[8];
  float s = 0.0f;
  for (int i = threadIdx.x; i < n; i += 256) s += partial[i];
  #pragma unroll
  for (int off = 16; off > 0; off >>= 1)
    s += __shfl_down(s, off, 32);
  if ((threadIdx.x & 31) == 0) wsum[threadIdx.x >> 5] = s;
  __syncthreads();
  if (threadIdx.x == 0) {
    float t = 0.0f;
    #pragma unroll
    for (int w = 0; w < 8; ++w) t += wsum[w];
    out[0] = t * INV_N;   // mean(res_x^2) + mean(res_y^2)
  }
}

extern "C" void kernel_launch(void* const* d_in, const int* in_sizes, int n_in,
                              void* d_out, int out_size, void* d_ws, size_t ws_size,
                              hipStream_t stream) {
  (void)in_sizes; (void)n_in; (void)out_size; (void)ws_size;
  const float* u = (const float*)d_in[0];
  const float* v = (const float*)d_in[1];
  float* out     = (float*)d_out;
  float* partial = (float*)d_ws;          // needs NBLK*4 = 16 KB of scratch

  ns_residual_kernel<<<NBLK, 256, 0, stream>>>(u, v, partial);
  final_reduce_kernel<<<1, 256, 0, stream>>>(partial, out, NBLK);
}